// FitTorch_283467842216
// MI455X (gfx1250) — compile-verified
//
#include <hip/hip_runtime.h>
#include <math.h>

typedef __attribute__((ext_vector_type(2))) float v2f;
typedef __attribute__((ext_vector_type(8))) float v8f;

#define ND      30
#define NDP     32
#define NATOMS  100000
#define NPAIRS  2000000
#define NCONF   50

// ---------------------------------------------------------------------------
// helpers
// ---------------------------------------------------------------------------
__device__ __forceinline__ float softplus_f(float z) {
    return (z > 20.f) ? z : log1pf(__expf(z));
}
__device__ __forceinline__ float sigmoid_f(float z) {
    return 1.f / (1.f + __expf(-z));
}

// One 16x32 = (16x32)x(32x32) GEMM-accumulate per wave using fp32 WMMA.
// A: 8 fragments (16x4 each).  Bm: 32x32 matrix in LDS, row-major (B[k][n]).
// C: two 16x16 f32 accumulator tiles (N-tiles t=0,1).
__device__ __forceinline__ void wave_mm(const v2f A[8], const float* __restrict__ Bm,
                                        v8f C[2], int nlo, int half) {
#pragma unroll
    for (int t = 0; t < 2; ++t) {
#pragma unroll
        for (int j = 0; j < 8; ++j) {
            const int kb = 4 * j + 2 * half;   // K base for this lane-half
            v2f b;
            b.x = Bm[kb * NDP + t * 16 + nlo];        // B[kb  ][16t+nlo]
            b.y = Bm[(kb + 1) * NDP + t * 16 + nlo];  // B[kb+1][16t+nlo]
            C[t] = __builtin_amdgcn_wmma_f32_16x16x4_f32(
                false, A[j], false, b, (short)0, C[t], false, false);
        }
    }
}

// Convert C-layout (row striped across lanes) -> A-layout fragments via LDS.
// stage: per-wave 16x32 float scratch.
__device__ __forceinline__ void c_to_a(const v8f C[2], v2f A[8],
                                       float* __restrict__ stage,
                                       int nlo, int half) {
    __syncthreads();
#pragma unroll
    for (int t = 0; t < 2; ++t)
#pragma unroll
        for (int r = 0; r < 8; ++r)
            stage[(r + 8 * half) * NDP + (16 * t + nlo)] = C[t][r];
    __syncthreads();
#pragma unroll
    for (int j = 0; j < 8; ++j) {
        const int kb = 4 * j + 2 * half;
        v2f a;
        a.x = stage[nlo * NDP + kb];
        a.y = stage[nlo * NDP + kb + 1];
        A[j] = a;
    }
}

// ---------------------------------------------------------------------------
// kernel 0: zero the output buffer (energy + forces)
// ---------------------------------------------------------------------------
__global__ void zero_kernel(float* __restrict__ out, int n) {
    int i = blockIdx.x * blockDim.x + threadIdx.x;
    if (i < n) out[i] = 0.f;
}

// ---------------------------------------------------------------------------
// kernel 1: per-atom MLP forward + input gradient (WMMA fp32)
// block = 256 threads = 8 waves; each wave handles 16 atoms.
// ---------------------------------------------------------------------------
__global__ __launch_bounds__(256) void mlp_grad_kernel(
    const float* __restrict__ x, const int* __restrict__ indices,
    const float* __restrict__ W0, const float* __restrict__ b0,
    const float* __restrict__ W1, const float* __restrict__ b1,
    const float* __restrict__ W2, const float* __restrict__ b2,
    const float* __restrict__ W3, const float* __restrict__ b3,
    float* __restrict__ energy_out, float* __restrict__ dEdD)
{
    __shared__ float sWT[3][NDP * NDP];   // forward:  B[k][n] = W[n][k]  (x @ W^T)
    __shared__ float sW [3][NDP * NDP];   // backward: B[k][n] = W[k][n]  (d @ W)
    __shared__ float sB [3][NDP];
    __shared__ float sW3[NDP];
    __shared__ float sStage[8][16 * NDP]; // per-wave C->A staging

    const int tid = threadIdx.x;

    // cooperative fill of padded weight matrices
    for (int idx = tid; idx < 3 * NDP * NDP; idx += 256) {
        const int m = idx / (NDP * NDP);
        const int e = idx % (NDP * NDP);
        const int k = e >> 5, n = e & 31;
        const float* W = (m == 0) ? W0 : (m == 1) ? W1 : W2;
        float wt = 0.f, wf = 0.f;
        if (k < ND && n < ND) { wt = W[n * ND + k]; wf = W[k * ND + n]; }
        sWT[m][e] = wt;
        sW [m][e] = wf;
    }
    if (tid < NDP) {
        sB[0][tid] = (tid < ND) ? b0[tid] : 0.f;
        sB[1][tid] = (tid < ND) ? b1[tid] : 0.f;
        sB[2][tid] = (tid < ND) ? b2[tid] : 0.f;
        sW3[tid]   = (tid < ND) ? W3[tid] : 0.f;
    }
    __syncthreads();

    const int wave = tid >> 5;
    const int lane = tid & 31;
    const int nlo  = lane & 15;
    const int half = lane >> 4;
    float* stage = sStage[wave];

    const int atomBase = blockIdx.x * 128 + wave * 16;

    // ---- A fragments from global x (16 atoms x 32 features, zero padded)
    v2f A0[8];
    {
        const int atom = atomBase + nlo;
        const int arow = (atom < NATOMS) ? atom : 0;
        const float* xr = x + (size_t)arow * ND;
#pragma unroll
        for (int j = 0; j < 8; ++j) {
            const int kb = 4 * j + 2 * half;
            v2f a;
            a.x = (kb     < ND) ? xr[kb]     : 0.f;
            a.y = (kb + 1 < ND) ? xr[kb + 1] : 0.f;
            A0[j] = a;
        }
    }

    // ---- forward: h0 = x @ W0^T + b0   (linear)
    v8f H0[2];
#pragma unroll
    for (int t = 0; t < 2; ++t) {
        const float bv = sB[0][t * 16 + nlo];
#pragma unroll
        for (int r = 0; r < 8; ++r) H0[t][r] = bv;
    }
    wave_mm(A0, sWT[0], H0, nlo, half);

    // ---- z1 = h0 @ W1^T + b1 ; h1 = softplus(z1), s1 = sigmoid(z1)
    v2f A1[8];
    c_to_a(H0, A1, stage, nlo, half);
    v8f Z1[2];
#pragma unroll
    for (int t = 0; t < 2; ++t) {
        const float bv = sB[1][t * 16 + nlo];
#pragma unroll
        for (int r = 0; r < 8; ++r) Z1[t][r] = bv;
    }
    wave_mm(A1, sWT[1], Z1, nlo, half);
    v8f H1[2], S1[2];
#pragma unroll
    for (int t = 0; t < 2; ++t)
#pragma unroll
        for (int r = 0; r < 8; ++r) {
            const float z = Z1[t][r];
            H1[t][r] = softplus_f(z);
            S1[t][r] = sigmoid_f(z);
        }

    // ---- z2 = h1 @ W2^T + b2 ; h2 = softplus(z2), s2 = sigmoid(z2)
    v2f A2[8];
    c_to_a(H1, A2, stage, nlo, half);
    v8f Z2[2];
#pragma unroll
    for (int t = 0; t < 2; ++t) {
        const float bv = sB[2][t * 16 + nlo];
#pragma unroll
        for (int r = 0; r < 8; ++r) Z2[t][r] = bv;
    }
    wave_mm(A2, sWT[2], Z2, nlo, half);
    v8f H2[2], S2[2];
#pragma unroll
    for (int t = 0; t < 2; ++t)
#pragma unroll
        for (int r = 0; r < 8; ++r) {
            const float z = Z2[t][r];
            H2[t][r] = softplus_f(z);
            S2[t][r] = sigmoid_f(z);
        }

    // ---- e_atom = h2 @ W3^T + b3 ; atomic add into per-config energy
    __syncthreads();
#pragma unroll
    for (int t = 0; t < 2; ++t)
#pragma unroll
        for (int r = 0; r < 8; ++r)
            stage[(r + 8 * half) * NDP + (16 * t + nlo)] = H2[t][r];
    __syncthreads();
    if (lane < 16) {
        const int atom = atomBase + lane;
        if (atom < NATOMS) {
            float e = b3[0];
#pragma unroll
            for (int n = 0; n < ND; ++n) e += stage[lane * NDP + n] * sW3[n];
            atomicAdd(&energy_out[indices[atom]], e);
        }
    }

    // ---- backward: dz2 = W3 * s2
    v8f D2[2];
#pragma unroll
    for (int t = 0; t < 2; ++t) {
        const float w3v = sW3[t * 16 + nlo];
#pragma unroll
        for (int r = 0; r < 8; ++r) D2[t][r] = w3v * S2[t][r];
    }

    // dh1 = dz2 @ W2 ; dz1 = dh1 * s1
    v2f Ad2[8];
    c_to_a(D2, Ad2, stage, nlo, half);
    v8f D1[2];
#pragma unroll
    for (int t = 0; t < 2; ++t)
#pragma unroll
        for (int r = 0; r < 8; ++r) D1[t][r] = 0.f;
    wave_mm(Ad2, sW[2], D1, nlo, half);
#pragma unroll
    for (int t = 0; t < 2; ++t)
#pragma unroll
        for (int r = 0; r < 8; ++r) D1[t][r] *= S1[t][r];

    // dh0 = dz1 @ W1  (layer0 is linear, so dz0 = dh0)
    v2f Ad1[8];
    c_to_a(D1, Ad1, stage, nlo, half);
    v8f D0[2];
#pragma unroll
    for (int t = 0; t < 2; ++t)
#pragma unroll
        for (int r = 0; r < 8; ++r) D0[t][r] = 0.f;
    wave_mm(Ad1, sW[1], D0, nlo, half);

    // dx = dz0 @ W0
    v2f Ad0[8];
    c_to_a(D0, Ad0, stage, nlo, half);
    v8f DX[2];
#pragma unroll
    for (int t = 0; t < 2; ++t)
#pragma unroll
        for (int r = 0; r < 8; ++r) DX[t][r] = 0.f;
    wave_mm(Ad0, sW[0], DX, nlo, half);

    // store dEdD (mask padding + tail atoms)
#pragma unroll
    for (int t = 0; t < 2; ++t) {
        const int n = 16 * t + nlo;
        if (n < ND) {
#pragma unroll
            for (int r = 0; r < 8; ++r) {
                const int atom = atomBase + r + 8 * half;
                if (atom < NATOMS) dEdD[(size_t)atom * ND + n] = DX[t][r];
            }
        }
    }
}

// ---------------------------------------------------------------------------
// kernel 2: pair streaming + force scatter (memory-bound: 720 MB of xd)
// ---------------------------------------------------------------------------
__global__ __launch_bounds__(256) void pair_force_kernel(
    const float* __restrict__ xd, const int* __restrict__ fidx,
    const float* __restrict__ dEdD, float* __restrict__ forces)
{
    const int p = blockIdx.x * 256 + threadIdx.x;
    if (p >= NPAIRS) return;

    const int neigh = fidx[(size_t)9 * p];       // force_indices[3p][0]
    const int aidx  = fidx[(size_t)9 * p + 1];   // force_indices[3p][1]

    const float* g = dEdD + (size_t)neigh * ND;  // L2-resident gather (temporal)
    const float* r = xd + (size_t)p * 90;        // streamed once (non-temporal)

    float c0 = 0.f, c1 = 0.f, c2 = 0.f;
#pragma unroll
    for (int i = 0; i < ND; i += 2) {
        v2f gv = *(const v2f*)(g + i);
        v2f a  = __builtin_nontemporal_load((const v2f*)(r + i));
        v2f b  = __builtin_nontemporal_load((const v2f*)(r + 30 + i));
        v2f c  = __builtin_nontemporal_load((const v2f*)(r + 60 + i));
        c0 += a.x * gv.x + a.y * gv.y;
        c1 += b.x * gv.x + b.y * gv.y;
        c2 += c.x * gv.x + c.y * gv.y;
    }
    atomicAdd(&forces[(size_t)aidx * 3 + 0], c0);
    atomicAdd(&forces[(size_t)aidx * 3 + 1], c1);
    atomicAdd(&forces[(size_t)aidx * 3 + 2], c2);
}

// ---------------------------------------------------------------------------
extern "C" void kernel_launch(void* const* d_in, const int* in_sizes, int n_in,
                              void* d_out, int out_size, void* d_ws, size_t ws_size,
                              hipStream_t stream) {
    const float* x       = (const float*)d_in[0];
    const float* xd      = (const float*)d_in[1];
    const int*   indices = (const int*)d_in[2];
    // d_in[3] atoms_per_structure: unused by the math
    const int*   fidx    = (const int*)d_in[4];
    const float* W0 = (const float*)d_in[5];
    const float* b0 = (const float*)d_in[6];
    const float* W1 = (const float*)d_in[7];
    const float* b1 = (const float*)d_in[8];
    const float* W2 = (const float*)d_in[9];
    const float* b2 = (const float*)d_in[10];
    const float* W3 = (const float*)d_in[11];
    const float* b3 = (const float*)d_in[12];

    float* out    = (float*)d_out;
    float* energy = out;            // [NCONF]
    float* forces = out + NCONF;    // [NATOMS*3]
    float* dEdD   = (float*)d_ws;   // [NATOMS*ND] = 12 MB scratch

    const int ntot = NCONF + NATOMS * 3;
    hipLaunchKernelGGL(zero_kernel, dim3((ntot + 255) / 256), dim3(256), 0, stream,
                       out, ntot);
    hipLaunchKernelGGL(mlp_grad_kernel, dim3((NATOMS + 127) / 128), dim3(256), 0, stream,
                       x, indices, W0, b0, W1, b1, W2, b2, W3, b3, energy, dEdD);
    hipLaunchKernelGGL(pair_force_kernel, dim3((NPAIRS + 255) / 256), dim3(256), 0, stream,
                       xd, fidx, dEdD, forces);
}